// LeakyIntegrator_90795608638292
// MI455X (gfx1250) — compile-verified
//
#include <hip/hip_runtime.h>
#include <cstdint>

// Problem constants (fixed by the reference harness).
namespace {
constexpr int T_DIM  = 4096;
constexpr int B_DIM  = 8192;
constexpr int CHUNKS = 32;               // chunks along T
constexpr int CLEN   = T_DIM / CHUNKS;   // 128 steps per chunk
constexpr int TILE   = 16;               // timesteps staged per async batch
constexpr int NTILES = CLEN / TILE;      // 8 tiles per chunk
constexpr int BLOCK  = 256;              // 8 wave32 per workgroup
}

// Low 32 bits of a generic pointer to LDS == wave-relative LDS byte offset.
__device__ __forceinline__ uint32_t lds_addr32(const void* p) {
  return (uint32_t)(uintptr_t)p;
}

// gfx1250 async copy: global -> LDS, 16B per lane, tracked by ASYNCcnt.
__device__ __forceinline__ void async_load_b128(uint32_t lds, uint64_t gaddr) {
  asm volatile("global_load_async_to_lds_b128 %0, %1, off"
               :: "v"(lds), "v"(gaddr)
               : "memory");
}

__device__ __forceinline__ void wait_async_prev_tile() {
  // <= 4 outstanding: the previous tile's 4 b128 sweeps have landed in LDS.
  asm volatile("s_wait_asynccnt 4" ::: "memory");
}
__device__ __forceinline__ void wait_async_all() {
  asm volatile("s_wait_asynccnt 0" ::: "memory");
}

// Cooperative per-wave staging of one TILE x BLOCK tile, 4 sweeps of b128.
// Lane l of wave w covers rows (l>>3)+4s, columns w*32 + (l&7)*4 .. +3, so a
// wave consumes only data it staged itself: s_wait_asynccnt is the only sync.
// gbase = &x[(t0 + rbase)*B + bcol]; lds_base already includes rbase/lane byte.
__device__ __forceinline__ void stage_tile(const float* gbase, uint32_t lds_base) {
  uint64_t ga = (uint64_t)(uintptr_t)gbase;
#pragma unroll
  for (int s = 0; s < 4; ++s) {
    async_load_b128(lds_base + (uint32_t)(s * 4 * BLOCK * 4), ga);
    ga += (uint64_t)4 * B_DIM * sizeof(float);   // advance 4 rows
  }
}

// ---------------------------------------------------------------------------
// Phase 1: per-(chunk, sequence) local scan with zero initial state.
// Emits the chunk's local carry L = (n10, n11, na0, na1) into ws planes.
// ---------------------------------------------------------------------------
__global__ __launch_bounds__(BLOCK)
void li_phase1(const float* __restrict__ x, const float* __restrict__ dptr,
               float* __restrict__ ws) {
  __shared__ float smem[2][TILE][BLOCK];

  const int tid = threadIdx.x;
  const int b   = blockIdx.x * BLOCK + tid;
  const int c   = blockIdx.y;
  const int t0  = c * CLEN;
  const float d = *dptr;

  // Staging geometry (per-lane within its wave).
  const int rbase = (tid >> 3) & 3;                 // first row this lane loads
  const int colc  = (tid & ~31) + ((tid & 7) << 2); // column within block
  const uint32_t laneb = (uint32_t)(colc * 4);      // byte offset within a row
  const uint32_t lb0 = lds_addr32(&smem[0][0][0]) + (uint32_t)(rbase * BLOCK * 4) + laneb;
  const uint32_t lb1 = lds_addr32(&smem[1][0][0]) + (uint32_t)(rbase * BLOCK * 4) + laneb;
  const float* gstage0 = x + (size_t)(t0 + rbase) * B_DIM + (blockIdx.x * BLOCK + colc);

  // prev at chunk start is an *input*: x[t0-1] (0 for the first chunk).
  float prev = (c == 0) ? 0.0f : x[(size_t)(t0 - 1) * B_DIM + b];
  float n10 = 0.f, n11 = 0.f, na0 = 0.f, na1 = 0.f;

  stage_tile(gstage0, lb0);
#pragma unroll 1
  for (int tile = 0; tile < NTILES; ++tile) {
    const int buf = tile & 1;
    if (tile + 1 < NTILES) {
      stage_tile(gstage0 + (size_t)(tile + 1) * TILE * B_DIM,
                 (tile + 1) & 1 ? lb1 : lb0);
      wait_async_prev_tile();
    } else {
      wait_async_all();
    }
#pragma unroll
    for (int r = 0; r < TILE; ++r) {
      const float xv  = smem[buf][r][tid];
      const float omp = 1.0f - prev;
      n10 = d * (n10 + xv * omp);
      n11 = d * (n11 + xv * prev);
      na0 = d * (na0 + omp);
      na1 = d * (na1 + prev);
      prev = xv;
    }
  }

  const size_t plane = (size_t)CHUNKS * B_DIM;
  const size_t idx   = (size_t)c * B_DIM + b;
  ws[0 * plane + idx] = n10;
  ws[1 * plane + idx] = n11;
  ws[2 * plane + idx] = na0;
  ws[3 * plane + idx] = na1;
}

// ---------------------------------------------------------------------------
// Phase 2: per-sequence scan over chunk carries.
// In-place: ws[c] is replaced by the chunk's true *initial* state S(c-1),
// using S(c) = L(c) + d^CLEN * S(c-1), S(-1) = 0.
// ---------------------------------------------------------------------------
__global__ __launch_bounds__(256)
void li_phase2(const float* __restrict__ dptr, float* __restrict__ ws) {
  const int b = blockIdx.x * 256 + threadIdx.x;
  const float d = *dptr;
  float dL = d;                 // d^128 via 7 squarings
#pragma unroll
  for (int i = 0; i < 7; ++i) dL *= dL;

  const size_t plane = (size_t)CHUNKS * B_DIM;
  float s0 = 0.f, s1 = 0.f, s2 = 0.f, s3 = 0.f;
#pragma unroll 1
  for (int c = 0; c < CHUNKS; ++c) {
    const size_t idx = (size_t)c * B_DIM + b;
    const float l0 = ws[0 * plane + idx];
    const float l1 = ws[1 * plane + idx];
    const float l2 = ws[2 * plane + idx];
    const float l3 = ws[3 * plane + idx];
    ws[0 * plane + idx] = s0;
    ws[1 * plane + idx] = s1;
    ws[2 * plane + idx] = s2;
    ws[3 * plane + idx] = s3;
    s0 = l0 + dL * s0;
    s1 = l1 + dL * s1;
    s2 = l2 + dL * s2;
    s3 = l3 + dL * s3;
  }
}

// ---------------------------------------------------------------------------
// Phase 3: rescan each chunk from its true initial state, emit predictions.
// x re-read should hit L2 (128 MB < 192 MB); output stores are non-temporal.
// ---------------------------------------------------------------------------
__global__ __launch_bounds__(BLOCK)
void li_phase3(const float* __restrict__ x, const float* __restrict__ dptr,
               const float* __restrict__ ws, float* __restrict__ out) {
  __shared__ float smem[2][TILE][BLOCK];

  const int tid = threadIdx.x;
  const int b   = blockIdx.x * BLOCK + tid;
  const int c   = blockIdx.y;
  const int t0  = c * CLEN;
  const float d = *dptr;

  const int rbase = (tid >> 3) & 3;
  const int colc  = (tid & ~31) + ((tid & 7) << 2);
  const uint32_t laneb = (uint32_t)(colc * 4);
  const uint32_t lb0 = lds_addr32(&smem[0][0][0]) + (uint32_t)(rbase * BLOCK * 4) + laneb;
  const uint32_t lb1 = lds_addr32(&smem[1][0][0]) + (uint32_t)(rbase * BLOCK * 4) + laneb;
  const float* gstage0 = x + (size_t)(t0 + rbase) * B_DIM + (blockIdx.x * BLOCK + colc);

  const size_t plane = (size_t)CHUNKS * B_DIM;
  const size_t sidx  = (size_t)c * B_DIM + b;
  float n10 = ws[0 * plane + sidx];
  float n11 = ws[1 * plane + sidx];
  float na0 = ws[2 * plane + sidx];
  float na1 = ws[3 * plane + sidx];
  float prev = (c == 0) ? 0.0f : x[(size_t)(t0 - 1) * B_DIM + b];

  stage_tile(gstage0, lb0);
#pragma unroll 1
  for (int tile = 0; tile < NTILES; ++tile) {
    const int buf = tile & 1;
    if (tile + 1 < NTILES) {
      stage_tile(gstage0 + (size_t)(tile + 1) * TILE * B_DIM,
                 (tile + 1) & 1 ? lb1 : lb0);
      wait_async_prev_tile();
    } else {
      wait_async_all();
    }
#pragma unroll
    for (int r = 0; r < TILE; ++r) {
      const float xv  = smem[buf][r][tid];
      const float omp = 1.0f - prev;
      n10 = d * (n10 + xv * omp);
      n11 = d * (n11 + xv * prev);
      na0 = d * (na0 + omp);
      na1 = d * (na1 + prev);
      // x is exactly 0.0 or 1.0, so pred = p0*(1-x)+p1*x reduces to a select;
      // only one division needed. Denominator >= 2, __fdividef is ~2 ulp.
      const bool  one = (xv != 0.0f);
      const float num = (one ? n11 : n10) + 1.0f;
      const float den = (one ? na1 : na0) + 2.0f;
      const float pred = __fdividef(num, den);
      const size_t t = (size_t)(t0 + tile * TILE + r);
      __builtin_nontemporal_store(pred, &out[t * B_DIM + b]);
      prev = xv;
    }
  }
}

// ---------------------------------------------------------------------------
extern "C" void kernel_launch(void* const* d_in, const int* in_sizes, int n_in,
                              void* d_out, int out_size, void* d_ws, size_t ws_size,
                              hipStream_t stream) {
  (void)in_sizes; (void)n_in; (void)out_size; (void)ws_size;
  const float* x    = (const float*)d_in[0];   // (T, B, 1) float32
  const float* dptr = (const float*)d_in[1];   // scalar decay_rate
  float* out = (float*)d_out;                  // (T, B, 1) float32
  float* ws  = (float*)d_ws;                   // 4 planes * CHUNKS * B floats = 4 MB

  dim3 grid(B_DIM / BLOCK, CHUNKS);            // (32, 32) blocks of 256
  li_phase1<<<grid, BLOCK, 0, stream>>>(x, dptr, ws);
  li_phase2<<<B_DIM / 256, 256, 0, stream>>>(dptr, ws);
  li_phase3<<<grid, BLOCK, 0, stream>>>(x, dptr, ws, out);
}